// GATv2Block_45578192945246
// MI455X (gfx1250) — compile-verified
//
#include <hip/hip_runtime.h>
#include <math.h>

#define N_NODES   50000
#define N_EDGES   800000
#define IN_CH     64
#define HEADS     4
#define OUT_CH    32
#define HC        128            // HEADS*OUT_CH
#define ET        (N_EDGES + N_NODES)
#define NEG_SLOPE 0.2f
#define BN_EPS    1e-5f
#define MEAN_BLOCKS 256

typedef float v2f __attribute__((ext_vector_type(2)));
typedef float v8f __attribute__((ext_vector_type(8)));

// ---------- helpers ----------
__device__ __forceinline__ int f2ord(float f) {
    int i = __float_as_int(f);
    return (i >= 0) ? i : (i ^ 0x7fffffff);
}
__device__ __forceinline__ float ord2f(int i) {
    return __int_as_float((i >= 0) ? i : (i ^ 0x7fffffff));
}

// ---------- 0: init accumulators (harness poisons buffers; we own zeroing) ----
__global__ __launch_bounds__(256)
void init_ws_kernel(float* __restrict__ out, float* __restrict__ denom,
                    int* __restrict__ nodeMax, float* __restrict__ colAcc)
{
    const int stride = gridDim.x * blockDim.x;
    int i = blockIdx.x * blockDim.x + threadIdx.x;
    for (int j = i; j < N_NODES * HC; j += stride) out[j] = 0.0f;
    for (int j = i; j < N_NODES * HEADS; j += stride) {
        denom[j]   = 0.0f;
        nodeMax[j] = (int)0x80000000;   // -inf in ordered-int space
    }
    for (int j = i; j < 2 * HC; j += stride) colAcc[j] = 0.0f;
}

// ---------- 1: xl = x@W_l, xr = x@W_r via V_WMMA_F32_16X16X4_F32 -------------
// One wave computes one 16x16 tile. 3125 M-tiles x (8 N-tiles x 2 matrices).
__global__ __launch_bounds__(256)
void gemm_xl_xr_kernel(const float* __restrict__ X,
                       const float* __restrict__ Wl, const float* __restrict__ Wr,
                       float* __restrict__ XL, float* __restrict__ XR)
{
    const int wid  = (blockIdx.x * blockDim.x + threadIdx.x) >> 5;
    const int lane = threadIdx.x & 31;
    const int half = lane >> 4;      // 0: lanes 0-15, 1: lanes 16-31
    const int idx  = lane & 15;

    const int mtile = wid >> 4;      // 0..3124
    const int nsel  = wid & 15;      // 0..7 -> XL, 8..15 -> XR
    if (mtile >= 3125) return;       // wave-uniform guard (grid is exact)

    const float* W = (nsel < 8) ? Wl : Wr;
    float*       D = (nsel < 8) ? XL : XR;
    const int ntile = nsel & 7;

    // A 16x4 f32 layout: lane half h holds K = 2h+j in VGPR j, M = idx
    const float* arow = X + (size_t)(mtile * 16 + idx) * IN_CH;
    // B 4x16 f32 layout: lane half h holds K = 2h+j in VGPR j, N = idx
    const float* bcol = W + (size_t)(ntile * 16 + idx);

    v8f c = {};
#pragma unroll
    for (int k0 = 0; k0 < IN_CH; k0 += 4) {
        v2f a, b;
        a.x = arow[k0 + 2 * half + 0];
        a.y = arow[k0 + 2 * half + 1];
        b.x = bcol[(size_t)(k0 + 2 * half + 0) * HC];
        b.y = bcol[(size_t)(k0 + 2 * half + 1) * HC];
        c = __builtin_amdgcn_wmma_f32_16x16x4_f32(
                false, a, false, b, (short)0, c, false, false);
    }

    // C/D layout: VGPR v -> M = v + 8*half, N = idx
#pragma unroll
    for (int v = 0; v < 8; ++v) {
        const int row = mtile * 16 + v + 8 * half;
        const int col = ntile * 16 + idx;
        D[(size_t)row * HC + col] = c[v];
    }
}

// ---------- 2: mean of (pos[src]-pos[dst]) over real edges -------------------
__global__ __launch_bounds__(256)
void edge_mean_partial_kernel(const int* __restrict__ srcA, const int* __restrict__ dstA,
                              const float* __restrict__ pos, float* __restrict__ part)
{
    __shared__ float sm[3][256];
    float a0 = 0.f, a1 = 0.f, a2 = 0.f;
    for (int e = blockIdx.x * blockDim.x + threadIdx.x; e < N_EDGES;
         e += gridDim.x * blockDim.x) {
        const int s = srcA[e], d = dstA[e];
        a0 += pos[s * 3 + 0] - pos[d * 3 + 0];
        a1 += pos[s * 3 + 1] - pos[d * 3 + 1];
        a2 += pos[s * 3 + 2] - pos[d * 3 + 2];
    }
    const int t = threadIdx.x;
    sm[0][t] = a0; sm[1][t] = a1; sm[2][t] = a2;
    __syncthreads();
    for (int o = 128; o > 0; o >>= 1) {
        if (t < o) { sm[0][t] += sm[0][t + o]; sm[1][t] += sm[1][t + o]; sm[2][t] += sm[2][t + o]; }
        __syncthreads();
    }
    if (t == 0) {
        part[blockIdx.x * 3 + 0] = sm[0][0];
        part[blockIdx.x * 3 + 1] = sm[1][0];
        part[blockIdx.x * 3 + 2] = sm[2][0];
    }
}

__global__ void edge_mean_final_kernel(const float* __restrict__ part, float* __restrict__ mean3)
{
    if (threadIdx.x < 3) {
        float s = 0.f;
        for (int b = 0; b < MEAN_BLOCKS; ++b) s += part[b * 3 + threadIdx.x];
        mean3[threadIdx.x] = s / (float)N_EDGES;
    }
}

// ---------- 3: per-edge GATv2 score + segment max ----------------------------
// One wave per edge; lane = channel (OUT_CH == 32 == wave32), loop over heads.
__global__ __launch_bounds__(256)
void edge_score_kernel(const int* __restrict__ srcA, const int* __restrict__ dstA,
                       const float* __restrict__ pos, const float* __restrict__ mean3,
                       const float* __restrict__ Wedge, const float* __restrict__ att,
                       const float* __restrict__ XL, const float* __restrict__ XR,
                       float* __restrict__ score, int* __restrict__ nodeMax)
{
    const int lane   = threadIdx.x & 31;
    const int waveId = (blockIdx.x * blockDim.x + threadIdx.x) >> 5;
    const int nWaves = (gridDim.x * blockDim.x) >> 5;

    for (int e = waveId; e < ET; e += nWaves) {
        int s, d; float ea0, ea1, ea2;
        if (e < N_EDGES) {
            s = srcA[e]; d = dstA[e];
            ea0 = pos[s * 3 + 0] - pos[d * 3 + 0];
            ea1 = pos[s * 3 + 1] - pos[d * 3 + 1];
            ea2 = pos[s * 3 + 2] - pos[d * 3 + 2];
        } else {
            s = d = e - N_EDGES;
            ea0 = mean3[0]; ea1 = mean3[1]; ea2 = mean3[2];
        }
        const float* xlrow = XL + (size_t)s * HC;
        const float* xrrow = XR + (size_t)d * HC;

        float sc[HEADS];
#pragma unroll
        for (int h = 0; h < HEADS; ++h) {
            const int off = h * OUT_CH + lane;
            float v = xlrow[off] + xrrow[off]
                    + ea0 * Wedge[0 * HC + off]
                    + ea1 * Wedge[1 * HC + off]
                    + ea2 * Wedge[2 * HC + off];
            v = (v > 0.f) ? v : NEG_SLOPE * v;          // LeakyReLU
            sc[h] = v * att[off];                       // att[h][c]
        }
#pragma unroll
        for (int h = 0; h < HEADS; ++h)
#pragma unroll
            for (int o = 16; o > 0; o >>= 1)
                sc[h] += __shfl_xor(sc[h], o, 32);

        if (lane < HEADS) {
            score[(size_t)e * HEADS + lane] = sc[lane];
            atomicMax(&nodeMax[d * HEADS + lane], f2ord(sc[lane]));
        }
    }
}

// ---------- 4: exp(score - max) and segment-sum denominator ------------------
__global__ __launch_bounds__(256)
void edge_exp_kernel(const int* __restrict__ dstA, const int* __restrict__ nodeMax,
                     float* __restrict__ score, float* __restrict__ denom)
{
    const int stride = gridDim.x * blockDim.x;
    for (int i = blockIdx.x * blockDim.x + threadIdx.x; i < ET * HEADS; i += stride) {
        const int e = i >> 2, h = i & 3;
        const int d = (e < N_EDGES) ? dstA[e] : (e - N_EDGES);
        const float m  = ord2f(nodeMax[d * HEADS + h]);
        const float sv = expf(score[i] - m);
        score[i] = sv;
        atomicAdd(&denom[d * HEADS + h], sv);
    }
}

// ---------- 5: out[dst] += alpha * xl[src] -----------------------------------
__global__ __launch_bounds__(256)
void edge_agg_kernel(const int* __restrict__ srcA, const int* __restrict__ dstA,
                     const float* __restrict__ XL, const float* __restrict__ score,
                     const float* __restrict__ denom, float* __restrict__ out)
{
    const int lane   = threadIdx.x & 31;
    const int waveId = (blockIdx.x * blockDim.x + threadIdx.x) >> 5;
    const int nWaves = (gridDim.x * blockDim.x) >> 5;

    for (int e = waveId; e < ET; e += nWaves) {
        int s, d;
        if (e < N_EDGES) { s = srcA[e]; d = dstA[e]; }
        else             { s = d = e - N_EDGES; }
#pragma unroll
        for (int h = 0; h < HEADS; ++h) {
            const float alpha = score[(size_t)e * HEADS + h]
                              / (denom[d * HEADS + h] + 1e-16f);
            const int off = h * OUT_CH + lane;
            atomicAdd(&out[(size_t)d * HC + off], alpha * XL[(size_t)s * HC + off]);
        }
    }
}

// ---------- 6: BatchNorm column statistics -----------------------------------
__global__ __launch_bounds__(128)
void bn_stats_kernel(const float* __restrict__ out, float* __restrict__ colAcc)
{
    const int c = threadIdx.x;                 // 0..127
    float s = 0.f, s2 = 0.f;
    for (int r = blockIdx.x; r < N_NODES; r += gridDim.x) {
        const float v = out[(size_t)r * HC + c];
        s += v; s2 += v * v;
    }
    atomicAdd(&colAcc[c],       s);
    atomicAdd(&colAcc[HC + c],  s2);
}

__global__ void bn_final_kernel(const float* __restrict__ colAcc,
                                const float* __restrict__ gamma,
                                const float* __restrict__ beta,
                                float* __restrict__ scaleShift)
{
    const int c = threadIdx.x;
    if (c < HC) {
        const float mu  = colAcc[c]      * (1.0f / (float)N_NODES);
        const float var = colAcc[HC + c] * (1.0f / (float)N_NODES) - mu * mu;
        const float sc  = gamma[c] * rsqrtf(var + BN_EPS);
        scaleShift[c]      = sc;
        scaleShift[HC + c] = beta[c] - mu * sc;
    }
}

__global__ __launch_bounds__(256)
void bn_apply_kernel(float* __restrict__ out, const float* __restrict__ scaleShift)
{
    const int stride = gridDim.x * blockDim.x;
    for (int i = blockIdx.x * blockDim.x + threadIdx.x; i < N_NODES * HC; i += stride) {
        const int c = i & (HC - 1);
        out[i] = out[i] * scaleShift[c] + scaleShift[HC + c];
    }
}

// ---------- host launcher ----------------------------------------------------
extern "C" void kernel_launch(void* const* d_in, const int* in_sizes, int n_in,
                              void* d_out, int out_size, void* d_ws, size_t ws_size,
                              hipStream_t stream)
{
    const float* x     = (const float*)d_in[0];   // [50000,64]
    const float* pos   = (const float*)d_in[1];   // [50000,3]
    const int*   eidx  = (const int*)  d_in[2];   // [2,800000]
    const float* W_l   = (const float*)d_in[3];   // [64,128]
    const float* W_r   = (const float*)d_in[4];   // [64,128]
    const float* W_e   = (const float*)d_in[5];   // [3,128]
    const float* att   = (const float*)d_in[6];   // [4,32]
    const float* gamma = (const float*)d_in[7];   // [128]
    const float* beta  = (const float*)d_in[8];   // [128]
    float* out = (float*)d_out;                   // [50000,128]

    const int* srcA = eidx;
    const int* dstA = eidx + N_EDGES;

    // workspace layout (floats)
    float* ws = (float*)d_ws;
    size_t off = 0;
    float* XL         = ws + off; off += (size_t)N_NODES * HC;     // 6.4M
    float* XR         = ws + off; off += (size_t)N_NODES * HC;     // 6.4M
    float* score      = ws + off; off += (size_t)ET * HEADS;       // 3.4M
    float* denom      = ws + off; off += (size_t)N_NODES * HEADS;  // 200K
    int*   nodeMax    = (int*)(ws + off); off += (size_t)N_NODES * HEADS;
    float* colAcc     = ws + off; off += 2 * HC;                   // sums + sumsq
    float* scaleShift = ws + off; off += 2 * HC;
    float* meanPart   = ws + off; off += 3 * MEAN_BLOCKS;
    float* mean3      = ws + off; off += 4;
    (void)ws_size; (void)n_in; (void)in_sizes; (void)out_size;

    // 0. zero/seed accumulators (d_out + denom + nodeMax + colAcc)
    init_ws_kernel<<<1024, 256, 0, stream>>>(out, denom, nodeMax, colAcc);

    // 1. xl / xr GEMMs (WMMA): 3125*16 tiles, 8 waves per 256-thread block
    gemm_xl_xr_kernel<<<6250, 256, 0, stream>>>(x, W_l, W_r, XL, XR);

    // 2. mean edge attribute for self-loops
    edge_mean_partial_kernel<<<MEAN_BLOCKS, 256, 0, stream>>>(srcA, dstA, pos, meanPart);
    edge_mean_final_kernel<<<1, 64, 0, stream>>>(meanPart, mean3);

    // 3. per-edge scores + segment max
    edge_score_kernel<<<2048, 256, 0, stream>>>(srcA, dstA, pos, mean3, W_e, att,
                                                XL, XR, score, nodeMax);

    // 4. softmax numerator + denominator
    edge_exp_kernel<<<4096, 256, 0, stream>>>(dstA, nodeMax, score, denom);

    // 5. weighted aggregation into out
    edge_agg_kernel<<<2048, 256, 0, stream>>>(srcA, dstA, XL, score, denom, out);

    // 6. BatchNorm (training-mode batch stats) applied in place
    bn_stats_kernel<<<1024, 128, 0, stream>>>(out, colAcc);
    bn_final_kernel<<<1, 128, 0, stream>>>(colAcc, gamma, beta, scaleShift);
    bn_apply_kernel<<<4096, 256, 0, stream>>>(out, scaleShift);
}